// ConvGlobalLocalCapsuleLayer_36000415875216
// MI455X (gfx1250) — compile-verified
//
#include <hip/hip_runtime.h>
#include <hip/hip_bf16.h>
#include <cstddef>
#include <cstdint>

typedef __attribute__((ext_vector_type(2))) float        v2f;
typedef __attribute__((ext_vector_type(8))) float        v8f;
typedef __attribute__((ext_vector_type(4))) unsigned int v4u;
typedef __attribute__((ext_vector_type(8))) int          v8i;
typedef __attribute__((ext_vector_type(4))) int          v4i;

#define NUM_CAPS   8    // output capsules
#define IN_CAPS    8    // input capsules
#define FILT       16   // filters per capsule
#define HH         32
#define WW         32
#define NOUT       128  // NUM_CAPS * FILT
#define WSLICE     (9 * FILT * NOUT)          // 18432 floats = one capsule's weights

// -----------------------------------------------------------------------------
// Kernel 1: implicit-GEMM conv producing votes[P][i][c*16+f] via fp32 WMMA.
// Block = 256 threads = 8 waves, ALL waves share one input capsule (cap =
// blockIdx.x & 7); wave w handles 16-pixel tile (blockIdx.x>>3)*8 + w.
// The capsule's full 72 KB weight panel (contiguous in memory) is DMA'd into
// LDS once per block by the Tensor Data Mover, then B operands are ds_loads.
// Per wave: M=16, N=128 (8 n-tiles), K=144 (36 steps of k=4).
// -----------------------------------------------------------------------------
__global__ __launch_bounds__(256) void caps_conv_votes_kernel(
    const float* __restrict__ x,      // [B,H,W,IN_CAPS,FILT]
    const float* __restrict__ Wt,     // [IN_CAPS,3,3,FILT,NOUT]
    float* __restrict__ votes)        // [P=32768][IN_CAPS][NOUT]
{
    extern __shared__ float ldsw[];             // 18432 floats (73728 B), offset 0

    const int lane   = threadIdx.x & 31;
    const int warp   = threadIdx.x >> 5;        // 0..7
    const int cap    = blockIdx.x & 7;          // shared by all waves in block
    const int tgroup = blockIdx.x >> 3;         // 0..255
    const int tile   = tgroup * 8 + warp;       // 16-pixel tile id
    const int P0     = tile << 4;
    const int bidx   = P0 >> 10;                // batch
    const int rem    = P0 & 1023;
    const int h      = rem >> 5;                // row
    const int w0     = rem & 31;                // 0 or 16

    const int m    = lane & 15;                 // M index (pixel in tile)
    const int koff = (lane >> 4) << 1;          // 0 or 2 : K sub-offset (ISA A layout)
    const int n_lo = lane & 15;                 // N index within n-tile

    // ---- TDM: DMA this capsule's 18432-float weight panel into LDS ----------
    if (threadIdx.x < 32) {                     // wave 0 issues the descriptor
        const uint64_t ga = (uint64_t)(uintptr_t)(Wt + (size_t)cap * WSLICE);
        v4u g0;
        g0.x = 1u;                                              // count=1 valid D#
        g0.y = 0u;                                              // lds_addr = 0
        g0.z = (uint32_t)(ga & 0xffffffffu);                    // global_addr[31:0]
        g0.w = (uint32_t)((ga >> 32) & 0x01ffffffu) | (2u << 30); // addr hi | type=2
        const v8i g1 = {
            0x00020000,                    // data_size=2 (4 bytes), no flags
            (int)(18432u << 16),           // tensor_dim0[15:0] in bits 63:48
            (int)(1u << 16),               // tensor_dim0 hi=0 ; tensor_dim1=1
            (int)(18432u << 16),           // tensor_dim1 hi=0 ; tile_dim0=18432
            1,                             // tile_dim1=1 ; tile_dim2=0
            18432,                         // tensor_dim0_stride[31:0]
            0,                             // stride hi ; tensor_dim1_stride lo
            0                              // tensor_dim1_stride hi
        };
        const v4i gz4 = {0, 0, 0, 0};
        const v8i gz8 = {0, 0, 0, 0, 0, 0, 0, 0};
        __builtin_amdgcn_tensor_load_to_lds(g0, g1, gz4, gz4, gz8, 0);
        __builtin_amdgcn_s_wait_tensorcnt(0);
    }
    __syncthreads();                            // LDS weights visible to all waves

    const v8f vzero = {};
    v8f acc[8];
#pragma unroll
    for (int nt = 0; nt < 8; ++nt) acc[nt] = vzero;

    for (int kh = 0; kh < 3; ++kh) {
        const int  hy  = h + kh - 1;
        const bool hok = (hy >= 0) && (hy < HH);
        for (int kw = 0; kw < 3; ++kw) {
            const int  wx = w0 + m + kw - 1;
            const bool ok = hok && (wx >= 0) && (wx < WW);
            const float* xbase =
                x + ((((size_t)(bidx * HH + hy) * WW + wx) * IN_CAPS + cap) * FILT);
            const float* wslice = ldsw + (size_t)((kh * 3 + kw) * FILT) * NOUT;
            for (int cb = 0; cb < 4; ++cb) {    // 4 cin-blocks of k=4
                const int cin = (cb << 2) + koff;
                v2f a;
                if (ok) {
                    const float2 t = *(const float2*)(xbase + cin);
                    a.x = t.x; a.y = t.y;
                } else {
                    a.x = 0.0f; a.y = 0.0f;
                }
                const float* wrow = wslice + (size_t)cin * NOUT;
#pragma unroll
                for (int nt = 0; nt < 8; ++nt) {
                    v2f bm;
                    bm.x = wrow[nt * 16 + n_lo];          // K row cin   (LDS)
                    bm.y = wrow[NOUT + nt * 16 + n_lo];   // K row cin+1 (LDS)
                    acc[nt] = __builtin_amdgcn_wmma_f32_16x16x4_f32(
                        false, a, false, bm, (short)0, acc[nt], false, false);
                }
            }
        }
    }

    // C layout: VGPR r, lanes 0-15 -> M=r, lanes 16-31 -> M=r+8; N = n_lo.
    const int m_hi = (lane >> 4) << 3;          // 0 or 8
#pragma unroll
    for (int nt = 0; nt < 8; ++nt) {
#pragma unroll
        for (int r = 0; r < 8; ++r) {
            const int    mm  = r + m_hi;
            const size_t off = (((size_t)(P0 + mm) * IN_CAPS + cap) * NOUT)
                               + nt * 16 + n_lo;
            votes[off] = acc[nt][r];
        }
    }
}

// -----------------------------------------------------------------------------
// Kernel 2: dynamic routing (3 iterations), one wave32 per pixel.
// Lane l <-> (c = l>>2, fquad = l&3): 4 contiguous f values per lane.
// Softmax over c: butterfly shfl_xor masks 4,8,16.  Dots over f: masks 1,2.
// -----------------------------------------------------------------------------
__global__ __launch_bounds__(256) void caps_routing_kernel(
    const float* __restrict__ votes,  // [P][IN_CAPS][NOUT]
    const float* __restrict__ bias,   // [NUM_CAPS][FILT] = 128 floats
    float* __restrict__ out)          // [P][NOUT]
{
    const int lane = threadIdx.x & 31;
    const int wv   = threadIdx.x >> 5;
    const int P    = blockIdx.x * 8 + wv;
    const int c    = lane >> 2;
    const int fq   = lane & 3;

    // Load this lane's slice of the votes: 4 f-values for each input capsule.
    float4 v[IN_CAPS];
    const float4* vp = (const float4*)(votes + (size_t)P * (IN_CAPS * NOUT));
#pragma unroll
    for (int i = 0; i < IN_CAPS; ++i)
        v[i] = vp[i * 32 + c * 4 + fq];
    const float4 bb = ((const float4*)bias)[c * 4 + fq];

    float logits[IN_CAPS];
#pragma unroll
    for (int i = 0; i < IN_CAPS; ++i) logits[i] = 0.0f;

    float4 act = {0.f, 0.f, 0.f, 0.f};
    for (int it = 0; it < 3; ++it) {
        // route[i][c] = softmax_c(logits[i][:])
        float route[IN_CAPS];
#pragma unroll
        for (int i = 0; i < IN_CAPS; ++i) {
            float mx = logits[i];
            mx = fmaxf(mx, __shfl_xor(mx, 4, 32));
            mx = fmaxf(mx, __shfl_xor(mx, 8, 32));
            mx = fmaxf(mx, __shfl_xor(mx, 16, 32));
            const float e = __expf(logits[i] - mx);
            float s = e;
            s += __shfl_xor(s, 4, 32);
            s += __shfl_xor(s, 8, 32);
            s += __shfl_xor(s, 16, 32);
            route[i] = e / s;
        }
        // preactivate[c][f] = sum_i route[i]*votes[i][c][f] + bias[c][f]
        float4 pre = bb;
#pragma unroll
        for (int i = 0; i < IN_CAPS; ++i) {
            pre.x += route[i] * v[i].x;
            pre.y += route[i] * v[i].y;
            pre.z += route[i] * v[i].z;
            pre.w += route[i] * v[i].w;
        }
        // squash over f (16 values across 4 lanes of the quad)
        float s2 = pre.x * pre.x + pre.y * pre.y + pre.z * pre.z + pre.w * pre.w;
        s2 += __shfl_xor(s2, 1, 32);
        s2 += __shfl_xor(s2, 2, 32);
        const float scale = s2 / (1.0f + s2) / sqrtf(s2 + 1e-7f);
        act.x = scale * pre.x;
        act.y = scale * pre.y;
        act.z = scale * pre.z;
        act.w = scale * pre.w;
        if (it < 2) {
            // agreement: logits[i][c] += <votes[i][c][:], act[c][:]>
#pragma unroll
            for (int i = 0; i < IN_CAPS; ++i) {
                float d = v[i].x * act.x + v[i].y * act.y +
                          v[i].z * act.z + v[i].w * act.w;
                d += __shfl_xor(d, 1, 32);
                d += __shfl_xor(d, 2, 32);
                logits[i] += d;
            }
        }
    }
    ((float4*)out)[(size_t)P * 32 + c * 4 + fq] = act;
}

extern "C" void kernel_launch(void* const* d_in, const int* in_sizes, int n_in,
                              void* d_out, int out_size, void* d_ws, size_t ws_size,
                              hipStream_t stream) {
    const float* x    = (const float*)d_in[0];  // [32,32,32,8,16]
    const float* Wt   = (const float*)d_in[1];  // [8,3,3,16,128]
    const float* bias = (const float*)d_in[2];  // [1,1,8,16]
    float*       out  = (float*)d_out;          // [32,32,32,8,16]
    float*       vws  = (float*)d_ws;           // votes: 32768*8*128 f32 = 134 MB

    // 256 tile-groups x 8 capsules; 73728 B dynamic LDS = one capsule's weights.
    caps_conv_votes_kernel<<<2048, 256, WSLICE * sizeof(float), stream>>>(x, Wt, vws);
    // 32768 pixels, one wave each, 8 waves/block.
    caps_routing_kernel<<<4096, 256, 0, stream>>>(vws, bias, out);
}